// FFMModel_9612136809114
// MI455X (gfx1250) — compile-verified
//
#include <hip/hip_runtime.h>
#include <hip/hip_bf16.h>
#include <stdint.h>

typedef __attribute__((ext_vector_type(2))) float v2f;
typedef __attribute__((ext_vector_type(8))) float v8f;

#define N_CAT   26
#define N_NUM   13
#define CARD    50000
#define ROW_B   624          /* 156 floats * 4 bytes per ffm_emb row */
#define WAVES   8
#define SPW     16           /* samples per wave (rows of the WMMA) */

// One wave handles 16 samples. For each pair i<j, lane L (s=L&15, h=L>>4)
// loads A-block u = sub[s][i][4j+2h .. +1] and B-block v = sub[s][j][4i+2h .. +1].
// v_wmma_f32_16x16x4_f32 accumulates D[s][s] += dot4(u_s, v_s) = pair[s,i,j].
__global__ __launch_bounds__(WAVES * 32) void ffm_fused(
    const int*   __restrict__ xcat,
    const float* __restrict__ xnum,
    const float* __restrict__ clin,
    const float* __restrict__ femb,
    const float* __restrict__ numW,
    const float* __restrict__ numb,
    float*       __restrict__ out)
{
    __shared__ uint32_t offL[WAVES][N_CAT][SPW];   // ffm_emb row byte offsets
    __shared__ uint32_t offC[WAVES][N_CAT][SPW];   // cat_lin_emb byte offsets
    __shared__ float    diag[WAVES][8][32];        // accumulator spill for diag pick

    const int tid  = threadIdx.x;
    const int w    = tid >> 5;
    const int lane = tid & 31;
    const int s    = lane & 15;     // sample slot (M=N row/col of WMMA)
    const int h    = lane >> 4;     // K-half: h=0 -> K{0,1}, h=1 -> K{2,3}
    const int sg0  = blockIdx.x * (WAVES * SPW) + w * SPW;

    // Stage per-sample row offsets (26 fields x 16 samples) into this wave's LDS slab.
    for (int k = lane; k < N_CAT * SPW; k += 32) {
        const int f  = k >> 4;
        const int ss = k & 15;
        const uint32_t idx = (uint32_t)f * (uint32_t)CARD +
                             (uint32_t)xcat[(sg0 + ss) * N_CAT + f];
        offL[w][f][ss] = idx * (uint32_t)ROW_B;   // max ~811MB, fits u32
        offC[w][f][ss] = idx * 4u;
    }
    __syncthreads();

    const char* fb = (const char*)femb;
    v8f acc = {0.f, 0.f, 0.f, 0.f, 0.f, 0.f, 0.f, 0.f};
    const uint32_t colh = (uint32_t)(h * 8);      // byte offset of element 2h

    for (int i = 0; i < N_CAT - 1; ++i) {
        const uint32_t oi   = offL[w][i][s];
        if (i + 1 < N_CAT) __builtin_prefetch(fb + offL[w][i + 1][s], 0, 1);
        const uint32_t vcol = (uint32_t)(16 * i) + colh;      // (4i+2h)*4
        for (int j = i + 1; j < N_CAT; ++j) {
            const uint32_t oj = offL[w][j][s];
            v2f a = *(const v2f*)(fb + (oi + (uint32_t)(16 * j) + colh)); // u: row i, col 4j+2h
            v2f b = *(const v2f*)(fb + (oj + vcol));                      // v: row j, col 4i+2h
            acc = __builtin_amdgcn_wmma_f32_16x16x4_f32(
                /*neg_a=*/false, a, /*neg_b=*/false, b,
                /*c_mod=*/(short)0, acc, /*reuse_a=*/false, /*reuse_b=*/false);
        }
    }

    // Spill accumulators; diagonal element for sample s lives at
    // vgpr = s&7, lane = s + (s>=8 ? 16 : 0)  (16x16 f32 C/D layout).
#pragma unroll
    for (int v = 0; v < 8; ++v) diag[w][v][lane] = acc[v];
    __syncthreads();

    if (lane < SPW) {
        const int ds = lane;
        const float ffm = diag[w][ds & 7][ds + ((ds >> 3) << 4)];

        float lin = numb[0];
        const float* xn = xnum + (size_t)(sg0 + ds) * N_NUM;
#pragma unroll
        for (int n = 0; n < N_NUM; ++n) lin += xn[n] * numW[n];
        const char* cb = (const char*)clin;
#pragma unroll
        for (int f = 0; f < N_CAT; ++f) lin += *(const float*)(cb + offC[w][f][ds]);

        out[sg0 + ds] = lin + ffm;
    }
}

extern "C" void kernel_launch(void* const* d_in, const int* in_sizes, int n_in,
                              void* d_out, int out_size, void* d_ws, size_t ws_size,
                              hipStream_t stream) {
    const int*   xcat = (const int*)  d_in[0];
    const float* xnum = (const float*)d_in[1];
    const float* clin = (const float*)d_in[2];
    const float* femb = (const float*)d_in[3];
    const float* numW = (const float*)d_in[4];
    const float* numb = (const float*)d_in[5];
    float*       out  = (float*)d_out;

    const int B = in_sizes[0] / N_CAT;            // 16384
    const int blocks = B / (WAVES * SPW);         // 128 blocks of 256 threads
    ffm_fused<<<blocks, WAVES * 32, 0, stream>>>(xcat, xnum, clin, femb, numW, numb, out);
}